// PCloudConv3d_10763188043863
// MI455X (gfx1250) — compile-verified
//
#include <hip/hip_runtime.h>

typedef __attribute__((ext_vector_type(2))) float v2f;
typedef __attribute__((ext_vector_type(8))) float v8f;

#define N_PTS   65536
#define KNBR    32
#define CIN     64
#define COUT    64
#define NKERN   27
#define BN_EPS  1e-5f

#define PTS_PER_BLOCK 128
#define THREADS       256
#define A_STRIDE      68   // 64 + 4 pad: conflict-free A-fragment ds_load_b64

#define N_SLICES      16   // point slices for the stats partial reduction
#define PTS_PER_SLICE (N_PTS / N_SLICES)

// d_ws layout (floats):
//   [0 .. 63]    final per-channel sum
//   [64 .. 127]  final per-channel sumsq
//   [128 ..]     partials[slice][group][8]  (4 sums + 4 sumsqs), 256*8 floats

// ---------------------------------------------------------------------------
// Kernel 1: gather + spatial filter + WMMA GEMM + bias + ReLU -> x (pre-BN)
// ---------------------------------------------------------------------------
__global__ __launch_bounds__(THREADS)
void pconv_kernel(const float* __restrict__ inputs,      // [N, 64]
                  const float* __restrict__ sw,          // [27, 64, 1]
                  const float* __restrict__ dw,          // [64, 64]
                  const float* __restrict__ biases,      // [64]
                  const int*   __restrict__ nn_count,    // [N]
                  const int*   __restrict__ nn_index,    // [N, 32]
                  const int*   __restrict__ filt_index,  // [N, 32]
                  float*       __restrict__ xout)        // [N, 64] pre-BN
{
    __shared__ alignas(16) float lds_sw[NKERN * CIN];              // 6912 B
    __shared__ alignas(16) float lds_sp[PTS_PER_BLOCK * A_STRIDE]; // 34816 B
    __shared__ alignas(16) v2f   lds_bfrag[16 * 4 * 32];           // 16384 B

    const int tid   = threadIdx.x;
    const int pbase = blockIdx.x * PTS_PER_BLOCK;

    // ---- stage spatial weights ----
    for (int i = tid; i < NKERN * CIN; i += THREADS)
        lds_sw[i] = sw[i];

    // ---- stage depth weights in WMMA B-fragment order ----
    // frag f = kk*4 + nt; lane l; v2f = {dw[4kk+off][16nt+l%16], dw[4kk+off+1][...]}
    for (int i = tid; i < 16 * 4 * 32; i += THREADS) {
        const int lane = i & 31;
        const int f    = i >> 5;
        const int kk   = f >> 2;
        const int nt   = f & 3;
        const int off  = (lane < 16) ? 0 : 2;
        const int col  = 16 * nt + (lane & 15);
        const int row  = 4 * kk + off;
        v2f b;
        b.x = dw[row * COUT + col];
        b.y = dw[(row + 1) * COUT + col];
        lds_bfrag[i] = b;
    }
    __syncthreads();

    // ---- gather + spatial filter: thread -> (point, 32-channel half) ----
    {
        const int prel = tid >> 1;
        const int p    = pbase + prel;
        const int c0   = (tid & 1) * 32;

        float acc[32];
        #pragma unroll
        for (int c = 0; c < 32; ++c) acc[c] = 0.0f;

        const int cnt  = nn_count[p];
        const int base = p * KNBR;
        for (int k = 0; k < cnt; ++k) {
            const int nidx = nn_index[base + k];
            const int fidx = filt_index[base + k];
            const float4* inrow = (const float4*)(inputs + nidx * CIN + c0);
            const float4* swrow = (const float4*)(lds_sw + fidx * CIN + c0);
            #pragma unroll
            for (int q = 0; q < 8; ++q) {
                const float4 a = inrow[q];
                const float4 w = swrow[q];
                acc[4 * q + 0] += a.x * w.x;
                acc[4 * q + 1] += a.y * w.y;
                acc[4 * q + 2] += a.z * w.z;
                acc[4 * q + 3] += a.w * w.w;
            }
        }
        const float scale = 1.0f / (float)cnt;
        float* dst = lds_sp + prel * A_STRIDE + c0;
        #pragma unroll
        for (int c = 0; c < 32; ++c) dst[c] = acc[c] * scale;
    }
    __syncthreads();

    // ---- GEMM: each wave owns 16 points, 4 output tiles of 16x16 ----
    const int wave = tid >> 5;           // wave32
    const int lane = tid & 31;
    const int mrow = lane & 15;
    const int off  = (lane < 16) ? 0 : 2;

    v8f cacc0 = {}, cacc1 = {}, cacc2 = {}, cacc3 = {};
    const float* arow = lds_sp + (16 * wave + mrow) * A_STRIDE + off;

    #pragma unroll
    for (int kk = 0; kk < 16; ++kk) {
        const v2f a  = *(const v2f*)(arow + 4 * kk);
        const v2f b0 = lds_bfrag[(kk * 4 + 0) * 32 + lane];
        const v2f b1 = lds_bfrag[(kk * 4 + 1) * 32 + lane];
        const v2f b2 = lds_bfrag[(kk * 4 + 2) * 32 + lane];
        const v2f b3 = lds_bfrag[(kk * 4 + 3) * 32 + lane];
        cacc0 = __builtin_amdgcn_wmma_f32_16x16x4_f32(false, a, false, b0, (short)0, cacc0, false, false);
        cacc1 = __builtin_amdgcn_wmma_f32_16x16x4_f32(false, a, false, b1, (short)0, cacc1, false, false);
        cacc2 = __builtin_amdgcn_wmma_f32_16x16x4_f32(false, a, false, b2, (short)0, cacc2, false, false);
        cacc3 = __builtin_amdgcn_wmma_f32_16x16x4_f32(false, a, false, b3, (short)0, cacc3, false, false);
    }

    // ---- epilogue: bias + ReLU, store pre-BN x ----
    const int mhalf = (lane < 16) ? 0 : 8;
    v8f tiles[4] = {cacc0, cacc1, cacc2, cacc3};
    #pragma unroll
    for (int nt = 0; nt < 4; ++nt) {
        const int   c    = 16 * nt + mrow;
        const float bias = biases[c];
        #pragma unroll
        for (int r = 0; r < 8; ++r) {
            float x = tiles[nt][r] + bias;
            x = fmaxf(x, 0.0f);
            const int prow = pbase + 16 * wave + mhalf + r;
            xout[prow * COUT + c] = x;
        }
    }
}

// ---------------------------------------------------------------------------
// Kernel 2a: partial per-channel sum / sumsq
//   grid = N_SLICES * 16 blocks; block -> (slice, channel group of 4)
// ---------------------------------------------------------------------------
__global__ __launch_bounds__(256)
void stats_partial_kernel(const float* __restrict__ x,        // [N, 64]
                          float*       __restrict__ partials) // [slice][group][8]
{
    __shared__ float4 ssum[256];
    __shared__ float4 ssq[256];

    const int slice = blockIdx.x >> 4;
    const int g     = blockIdx.x & 15;     // channels 4g..4g+3
    const int t     = threadIdx.x;
    const int p0    = slice * PTS_PER_SLICE;

    float4 s = {0.f, 0.f, 0.f, 0.f};
    float4 q = {0.f, 0.f, 0.f, 0.f};
    for (int p = t; p < PTS_PER_SLICE; p += 256) {
        const float4 v = ((const float4*)x)[(p0 + p) * (COUT / 4) + g];
        s.x += v.x; s.y += v.y; s.z += v.z; s.w += v.w;
        q.x += v.x * v.x; q.y += v.y * v.y; q.z += v.z * v.z; q.w += v.w * v.w;
    }
    ssum[t] = s; ssq[t] = q;
    __syncthreads();

    for (int step = 128; step > 0; step >>= 1) {
        if (t < step) {
            float4 a = ssum[t], b = ssum[t + step];
            a.x += b.x; a.y += b.y; a.z += b.z; a.w += b.w;
            ssum[t] = a;
            float4 c = ssq[t], d = ssq[t + step];
            c.x += d.x; c.y += d.y; c.z += d.z; c.w += d.w;
            ssq[t] = c;
        }
        __syncthreads();
    }
    if (t == 0) {
        float* dst = partials + (slice * 16 + g) * 8;
        const float4 a = ssum[0];
        const float4 c = ssq[0];
        dst[0] = a.x; dst[1] = a.y; dst[2] = a.z; dst[3] = a.w;
        dst[4] = c.x; dst[5] = c.y; dst[6] = c.z; dst[7] = c.w;
    }
}

// ---------------------------------------------------------------------------
// Kernel 2b: final deterministic reduce over slices -> gstats[0..127]
// ---------------------------------------------------------------------------
__global__ __launch_bounds__(128)
void stats_final_kernel(const float* __restrict__ partials,
                        float*       __restrict__ gstats)
{
    const int t = threadIdx.x;            // t<64: sum of channel t; t>=64: sumsq
    if (t >= 2 * COUT) return;
    const int c    = t & 63;
    const int isq  = (t >= COUT) ? 4 : 0;
    const int g    = c >> 2;
    const int comp = c & 3;

    float acc = 0.f;
    #pragma unroll
    for (int slice = 0; slice < N_SLICES; ++slice)
        acc += partials[(slice * 16 + g) * 8 + isq + comp];
    gstats[t] = acc;
}

// ---------------------------------------------------------------------------
// Kernel 3: BatchNorm normalize in-place on d_out
// ---------------------------------------------------------------------------
__global__ __launch_bounds__(256)
void bn_kernel(const float* __restrict__ gstats,
               const float* __restrict__ gamma,
               const float* __restrict__ beta,
               float*       __restrict__ xio)
{
    const int i  = blockIdx.x * 256 + threadIdx.x;   // float4 index
    const int cb = (i & 15) * 4;                     // 16 float4 per 64-ch row
    const float4 x = ((const float4*)xio)[i];
    const float  xa[4] = {x.x, x.y, x.z, x.w};
    const float  invN  = 1.0f / (float)N_PTS;

    float o[4];
    #pragma unroll
    for (int j = 0; j < 4; ++j) {
        const int   c    = cb + j;
        const float mean = gstats[c] * invN;
        const float var  = gstats[COUT + c] * invN - mean * mean;
        o[j] = gamma[c] * (xa[j] - mean) * rsqrtf(var + BN_EPS) + beta[c];
    }
    const float4 r = {o[0], o[1], o[2], o[3]};
    ((float4*)xio)[i] = r;
}

// ---------------------------------------------------------------------------
extern "C" void kernel_launch(void* const* d_in, const int* in_sizes, int n_in,
                              void* d_out, int out_size, void* d_ws, size_t ws_size,
                              hipStream_t stream) {
    const float* inputs     = (const float*)d_in[0];
    const float* sw         = (const float*)d_in[1];
    const float* dw         = (const float*)d_in[2];
    const float* biases     = (const float*)d_in[3];
    const float* gamma      = (const float*)d_in[4];
    const float* beta       = (const float*)d_in[5];
    const int*   nn_count   = (const int*)d_in[6];
    const int*   nn_index   = (const int*)d_in[7];
    const int*   filt_index = (const int*)d_in[8];
    float*       out        = (float*)d_out;
    float*       gstats     = (float*)d_ws;          // [128]
    float*       partials   = (float*)d_ws + 128;    // [N_SLICES*16*8]

    pconv_kernel<<<N_PTS / PTS_PER_BLOCK, THREADS, 0, stream>>>(
        inputs, sw, dw, biases, nn_count, nn_index, filt_index, out);

    stats_partial_kernel<<<N_SLICES * 16, 256, 0, stream>>>(out, partials);

    stats_final_kernel<<<1, 128, 0, stream>>>(partials, gstats);

    bn_kernel<<<(N_PTS * COUT / 4) / 256, 256, 0, stream>>>(gstats, gamma, beta, out);
}